// LinearDecoderVarLen_25357486916301
// MI455X (gfx1250) — compile-verified
//
#include <hip/hip_runtime.h>
#include <hip/hip_bf16.h>

#define T_TOTAL 131072
#define E_DIM   2048
#define O_DIM   1024
#define B_SEG   16

#define ROWS_PER_BLOCK 256
#define THREADS_P1     256
#define COLS_PER_BLOCK (THREADS_P1 * 4)   // 1024 columns per block

typedef __attribute__((ext_vector_type(2))) float v2f;
typedef __attribute__((ext_vector_type(4))) float v4f;
typedef __attribute__((ext_vector_type(8))) float v8f;

// ---------------------------------------------------------------------------
// Pass 0: zero the segment-sum accumulator in workspace (harness poisons ws).
// ---------------------------------------------------------------------------
__global__ void zero_sums_kernel(float* __restrict__ sums) {
    int i = blockIdx.x * blockDim.x + threadIdx.x;
    if (i < B_SEG * E_DIM) sums[i] = 0.0f;
}

// ---------------------------------------------------------------------------
// Pass 1: streaming segmented sum over x (the HBM-bound 1.07 GB pass).
// Each thread owns 4 consecutive columns; each block owns 256 consecutive
// rows x 1024 columns. Register accumulation, atomic flush at segment edges.
// ---------------------------------------------------------------------------
__global__ __launch_bounds__(THREADS_P1)
void seg_sum_kernel(const float* __restrict__ x,
                    const int*   __restrict__ lengths,
                    float*       __restrict__ sums) {
    // Per-thread exclusive-scan of the 16 segment lengths (tiny, L2/scalar hit).
    int offs[B_SEG + 1];
    offs[0] = 0;
#pragma unroll
    for (int i = 0; i < B_SEG; ++i) offs[i + 1] = offs[i] + lengths[i];

    const int col = blockIdx.x * COLS_PER_BLOCK + threadIdx.x * 4;
    const int r0  = blockIdx.y * ROWS_PER_BLOCK;
    const int r1  = r0 + ROWS_PER_BLOCK;

    // Find the segment containing row r0.
    int s = 0;
    while (offs[s + 1] <= r0) ++s;

    const float* xp = x + (size_t)r0 * E_DIM + col;
    v4f acc = {0.f, 0.f, 0.f, 0.f};
    int row = r0;

    while (row < r1) {
        const int segEnd = (offs[s + 1] < r1) ? offs[s + 1] : r1;
#pragma unroll 4
        for (; row < segEnd; ++row) {
            // x is read exactly once and is far larger than L2: non-temporal.
            v4f v = __builtin_nontemporal_load((const v4f*)xp);
            acc.x += v.x; acc.y += v.y; acc.z += v.z; acc.w += v.w;
            xp += E_DIM;
        }
        // Flush this segment's partial sum (few atomics per thread total).
        float* dp = sums + (size_t)s * E_DIM + col;
        atomicAdd(dp + 0, acc.x);
        atomicAdd(dp + 1, acc.y);
        atomicAdd(dp + 2, acc.z);
        atomicAdd(dp + 3, acc.w);
        acc.x = 0.f; acc.y = 0.f; acc.z = 0.f; acc.w = 0.f;
        ++s;
    }
}

// ---------------------------------------------------------------------------
// Pass 2: out[16 x 1024] = (sums/len) @ W + b using V_WMMA_F32_16X16X4_F32.
// One wave per 16x16 output tile, K=2048 in steps of 4.
//
// A (16x4 f32) layout per ISA 7.12.2: lane M = lane&15;
//   VGPR0 holds K = {0 | lanes 0-15, 2 | lanes 16-31}, VGPR1 holds K+1.
// B (4x16 f32) mirrors: lane N = lane&15, same K-pair-per-lane-half split.
// C/D (16x16 f32, 8 VGPRs): VGPR r -> M = r (lanes 0-15) / r+8 (lanes 16-31),
//   N = lane&15.
// ---------------------------------------------------------------------------
__global__ __launch_bounds__(32)
void gemm_wmma_kernel(const float* __restrict__ sums,
                      const int*   __restrict__ lengths,
                      const float* __restrict__ W,
                      const float* __restrict__ bias,
                      float*       __restrict__ out) {
    const int lane  = threadIdx.x;          // 0..31, wave32
    const int mn    = lane & 15;            // A-row M and B-col N for this lane
    const int kk    = (lane >> 4) << 1;     // 0 for lanes 0-15, 2 for lanes 16-31
    const int ncol0 = blockIdx.x * 16;

    const float rlen = 1.0f / (float)lengths[mn];
    const float* arow = sums + (size_t)mn * E_DIM;
    const float* bcol = W + ncol0 + mn;

    v8f c = {0.f, 0.f, 0.f, 0.f, 0.f, 0.f, 0.f, 0.f};

#pragma unroll 4
    for (int k = 0; k < E_DIM; k += 4) {
        v2f av = *(const v2f*)(arow + k + kk);    // A[M][k+kk], A[M][k+kk+1]
        v2f a;
        a.x = av.x * rlen;
        a.y = av.y * rlen;

        v2f b;
        b.x = bcol[(size_t)(k + kk)     * O_DIM]; // W[k+kk  ][ncol0+N]
        b.y = bcol[(size_t)(k + kk + 1) * O_DIM]; // W[k+kk+1][ncol0+N]

        // 8 args: (neg_a, A, neg_b, B, c_mod, C, reuse_a, reuse_b)
        c = __builtin_amdgcn_wmma_f32_16x16x4_f32(
                false, a, false, b, (short)0, c, false, false);
    }

    const int mbase = (lane < 16) ? 0 : 8;
    const int ncol  = ncol0 + (lane & 15);
    const float bb  = bias[ncol];
#pragma unroll
    for (int r = 0; r < 8; ++r) {
        out[(size_t)(mbase + r) * O_DIM + ncol] = c[r] + bb;
    }
}

// ---------------------------------------------------------------------------
// Launch
// ---------------------------------------------------------------------------
extern "C" void kernel_launch(void* const* d_in, const int* in_sizes, int n_in,
                              void* d_out, int out_size, void* d_ws, size_t ws_size,
                              hipStream_t stream) {
    const float* x       = (const float*)d_in[0];   // [T, E]
    const float* W       = (const float*)d_in[1];   // [E, O]
    const float* b       = (const float*)d_in[2];   // [O]
    const int*   lengths = (const int*)  d_in[3];   // [B]
    float*       out     = (float*)d_out;           // [B, O]
    float*       sums    = (float*)d_ws;            // [B, E] scratch (128 KB)

    (void)in_sizes; (void)n_in; (void)out_size; (void)ws_size;

    // Pass 0: zero the accumulator.
    zero_sums_kernel<<<(B_SEG * E_DIM + 255) / 256, 256, 0, stream>>>(sums);

    // Pass 1: segmented sum (HBM-bound streaming pass).
    dim3 grid1(E_DIM / COLS_PER_BLOCK, T_TOTAL / ROWS_PER_BLOCK);  // (2, 512)
    seg_sum_kernel<<<grid1, THREADS_P1, 0, stream>>>(x, lengths, sums);

    // Pass 2: fp32 WMMA GEMM with fused mean-divide and bias.
    gemm_wmma_kernel<<<O_DIM / 16, 32, 0, stream>>>(sums, lengths, W, b, out);
}